// YOLOLoss_45904610459925
// MI455X (gfx1250) — compile-verified
//
#include <hip/hip_runtime.h>
#include <hip/hip_bf16.h>
#include <stdint.h>

// ---------------- problem constants ----------------
#define B_   16
#define A_   3
#define Hh   76
#define Ww   76
#define T_   50
#define C_   80
#define HW   (Hh * Ww)            // 5776
#define CELLS_IMG (A_ * HW)       // 17328
#define MAPN (B_ * A_ * HW)       // 277248
#define BLK  256
#define BLKS_X ((CELLS_IMG + BLK - 1) / BLK)   // 68
#define NPART (BLKS_X * B_)                     // 1088

// scaled anchors (ANCHORS / 8) and scaled total anchors (TOTAL / 8)
__device__ __forceinline__ float sa_w(int a) { return a == 0 ? 1.25f : (a == 1 ? 2.0f : 4.125f); }
__device__ __forceinline__ float sa_h(int a) { return a == 0 ? 1.625f : (a == 1 ? 3.75f : 2.875f); }

__device__ __forceinline__ float slog(float v) { return v > 0.0f ? logf(v) : -100.0f; }
__device__ __forceinline__ float bce(float p, float t) {
    return -(t * slog(p) + (1.0f - t) * slog(1.0f - p));
}
__device__ __forceinline__ float smooth_l1(float p, float t) {
    float d = fabsf(p - t);
    return d < 1.0f ? 0.5f * d * d : d - 0.5f;
}
__device__ __forceinline__ float sigmoidf(float x) { return 1.0f / (1.0f + expf(-x)); }

// ---------------- kernel 0: zero the scatter map + valid counter ----------------
__global__ void k_zero(int* __restrict__ p, int n) {
    int i = blockIdx.x * blockDim.x + threadIdx.x;
    if (i < n) p[i] = 0;
}

// ---------------- kernel 1: per-target preprocessing ----------------
// igbuf[b*T+t] = {gx,gy,gw,gh} (zeros when invalid)  -> used for ignore IoU
// rec  [b*T+t] = {tx,ty,tw,th,scale,cls}             -> used at masked cells
// map[cell]    = (t+1) of last (max-t) target writing that cell
__global__ void k_prep(const float* __restrict__ tgt,
                       int* __restrict__ map, int* __restrict__ nvalid,
                       float* __restrict__ igbuf, float* __restrict__ rec) {
    int idx = blockIdx.x * blockDim.x + threadIdx.x;
    if (idx >= B_ * T_) return;
    int b = idx / T_, t = idx - b * T_;

    const float* tr = tgt + (size_t)idx * 5;
    float cls = tr[0], cx = tr[1], cy = tr[2], tw = tr[3], th = tr[4];
    bool valid = (cls + cx + cy + tw + th) != 0.0f;

    float gx = cx * (float)Ww, gy = cy * (float)Hh;
    float gw = tw * (float)Ww, gh = th * (float)Hh;

    float* ig = igbuf + (size_t)idx * 4;
    ig[0] = valid ? gx : 0.0f;
    ig[1] = valid ? gy : 0.0f;
    ig[2] = valid ? gw : 0.0f;
    ig[3] = valid ? gh : 0.0f;

    if (!valid) return;
    atomicAdd(nvalid, 1);

    // anchor match against 9 scaled total anchors (first-max wins, like jnp.argmax)
    const float taw[9] = {1.25f, 2.0f, 4.125f, 3.75f, 7.75f, 7.375f, 14.5f, 19.5f, 46.625f};
    const float tah[9] = {1.625f, 3.75f, 2.875f, 7.625f, 5.625f, 14.875f, 11.25f, 24.75f, 40.75f};
    float best = -1.0f; int bi = 0;
    #pragma unroll
    for (int k = 0; k < 9; ++k) {
        float inter = fminf(gw, taw[k]) * fminf(gh, tah[k]);
        float iou = inter / (gw * gh + taw[k] * tah[k] - inter + 1e-16f);
        if (iou > best) { best = iou; bi = k; }
    }
    if (bi >= A_) return;                       // LUT: only anchors 0..2 belong to this head

    int gi = (int)gx, gj = (int)gy;
    if (gi < 0 || gi >= Ww || gj < 0 || gj >= Hh) return;   // scatter mode='drop'

    float* rr = rec + (size_t)idx * 6;
    rr[0] = gx - (float)gi;
    rr[1] = gy - (float)gj;
    rr[2] = logf(gw / sa_w(bi) + 1e-16f);
    rr[3] = logf(gh / sa_h(bi) + 1e-16f);
    rr[4] = 2.0f - tw * th;
    rr[5] = (float)(int)cls;

    int cell = ((b * A_ + bi) * Hh + gj) * Ww + gi;
    atomicMax(&map[cell], t + 1);               // deterministic duplicate resolution
}

// ---------------- kernel 2: main per-cell pass ----------------
__global__ void __launch_bounds__(BLK) k_main(const float* __restrict__ in,
                                              const int* __restrict__ map,
                                              const float* __restrict__ igbuf,
                                              const float* __restrict__ rec,
                                              float* __restrict__ partials) {
    __shared__ float4 s_tg[T_];
    __shared__ float red[BLK];

    const int b = blockIdx.y;
    const int tid = threadIdx.x;

    // CDNA5 async copy: stage this image's 50 gt boxes (16B each) into LDS.
    if (tid < T_) {
        uint64_t gaddr = (uint64_t)(uintptr_t)(igbuf + ((size_t)b * T_ + tid) * 4);
        uint32_t laddr = (uint32_t)(uintptr_t)(&s_tg[tid]);   // low 32 bits of flat = LDS offset
        asm volatile("global_load_async_to_lds_b128 %0, %1, off"
                     :: "v"(laddr), "v"(gaddr) : "memory");
    }
    asm volatile("s_wait_asynccnt 0" ::: "memory");
    __syncthreads();

    float acc[6] = {0.f, 0.f, 0.f, 0.f, 0.f, 0.f};

    const int c = blockIdx.x * BLK + tid;
    if (c < CELLS_IMG) {
        const int a = c / HW;
        const int r = c - a * HW;
        const int j = r / Ww;
        const int i = r - j * Ww;

        const size_t base = (size_t)(b * 255 + a * 85) * HW + (size_t)r;
        const float xr = in[base];
        const float yr = in[base + (size_t)HW];
        const float wr = in[base + (size_t)2 * HW];
        const float hr = in[base + (size_t)3 * HW];
        const float cr = in[base + (size_t)4 * HW];

        const float sx = sigmoidf(xr);
        const float sy = sigmoidf(yr);
        const float sc = sigmoidf(cr);

        const float pbx = sx + (float)i;
        const float pby = sy + (float)j;
        const float pbw = expf(wr) * sa_w(a);
        const float pbh = expf(hr) * sa_h(a);

        const float b1x1 = pbx - 0.5f * pbw, b1x2 = pbx + 0.5f * pbw;
        const float b1y1 = pby - 0.5f * pbh, b1y2 = pby + 0.5f * pbh;
        const float a1 = (b1x2 - b1x1) * (b1y2 - b1y1);

        bool ign = false;
        #pragma unroll 5
        for (int t = 0; t < T_; ++t) {
            float4 g = s_tg[t];     // invalid targets are zeros -> IoU == 0
            float b2x1 = g.x - 0.5f * g.z, b2x2 = g.x + 0.5f * g.z;
            float b2y1 = g.y - 0.5f * g.w, b2y2 = g.y + 0.5f * g.w;
            float iw = fmaxf(fminf(b1x2, b2x2) - fmaxf(b1x1, b2x1), 0.0f);
            float ih = fmaxf(fminf(b1y2, b2y2) - fmaxf(b1y1, b2y1), 0.0f);
            float inter = iw * ih;
            float a2 = g.z * g.w;
            float iou = inter / (a1 + a2 - inter + 1e-16f);
            ign = ign || (iou >= 0.5f);
        }

        const int t0 = map[(size_t)(b * A_ + a) * HW + r];
        if (t0 > 0) {
            // object cell: x/y/w/h + obj-conf + 80-class BCE (only here do we touch class channels)
            const float* rr = rec + ((size_t)b * T_ + (t0 - 1)) * 6;
            const float tx = rr[0], ty = rr[1], twt = rr[2], tht = rr[3];
            const float scale = rr[4];
            const int cls = (int)rr[5];
            acc[0] = scale * bce(sx, tx);
            acc[1] = scale * bce(sy, ty);
            acc[2] = scale * smooth_l1(wr, twt);
            acc[3] = scale * smooth_l1(hr, tht);
            acc[4] = -slog(sc);                          // bce(conf, 1)
            float s = 0.0f;
            const float* pc = in + base + (size_t)5 * HW;
            for (int cc = 0; cc < C_; ++cc) {
                float sp = sigmoidf(pc[(size_t)cc * HW]);
                s += (cc == cls) ? -slog(sp) : -slog(1.0f - sp);
            }
            acc[5] = s;
        } else if (!ign) {
            acc[4] = 0.5f * (-slog(1.0f - sc));          // 0.5 * bce(conf, 0) on noobj cells
        }
        // ignored & unmasked cells contribute nothing (noobj == 0, mask == 0)
    }

    // fixed-shape block reduction (deterministic) -> per-block partials
    const int blk = blockIdx.y * gridDim.x + blockIdx.x;
    #pragma unroll
    for (int cmp = 0; cmp < 6; ++cmp) {
        red[tid] = acc[cmp];
        __syncthreads();
        for (int st = BLK / 2; st > 0; st >>= 1) {
            if (tid < st) red[tid] += red[tid + st];
            __syncthreads();
        }
        if (tid == 0) partials[(size_t)blk * 6 + cmp] = red[0];
        __syncthreads();
    }
}

// ---------------- kernel 3: final deterministic reduction ----------------
__global__ void k_final(const float* __restrict__ partials,
                        const int* __restrict__ nvalid,
                        float* __restrict__ out) {
    __shared__ float red[BLK];
    __shared__ float comp_s[6];
    const int tid = threadIdx.x;
    for (int cmp = 0; cmp < 6; ++cmp) {
        float s = 0.0f;
        for (int k = tid; k < NPART; k += BLK) s += partials[(size_t)k * 6 + cmp];
        red[tid] = s;
        __syncthreads();
        for (int st = BLK / 2; st > 0; st >>= 1) {
            if (tid < st) red[tid] += red[tid + st];
            __syncthreads();
        }
        if (tid == 0) comp_s[cmp] = red[0];
        __syncthreads();
    }
    if (tid == 0) {
        float n = (float)(*nvalid);
        float lx = comp_s[0] / n, ly = comp_s[1] / n;
        float lw = comp_s[2] / n, lh = comp_s[3] / n;
        float lc = comp_s[4] / n, lcl = comp_s[5] / n;
        out[0] = 2.5f * (lx + ly) + 2.5f * (lw + lh) + lc + lcl;
        out[1] = lx; out[2] = ly; out[3] = lw; out[4] = lh; out[5] = lc; out[6] = lcl;
    }
}

// ---------------- launch ----------------
extern "C" void kernel_launch(void* const* d_in, const int* in_sizes, int n_in,
                              void* d_out, int out_size, void* d_ws, size_t ws_size,
                              hipStream_t stream) {
    (void)in_sizes; (void)n_in; (void)out_size; (void)ws_size;
    const float* input   = (const float*)d_in[0];
    const float* targets = (const float*)d_in[1];
    float* out = (float*)d_out;

    // workspace layout (igbuf kept 16B-aligned for the async b128 copy)
    int*   map      = (int*)d_ws;                 // MAPN ints
    int*   nvalid   = map + MAPN;                 // 1 int
    float* igbuf    = (float*)(map + MAPN + 4);   // B*T*4 floats
    float* rec      = igbuf + (size_t)B_ * T_ * 4; // B*T*6 floats
    float* partials = rec + (size_t)B_ * T_ * 6;   // NPART*6 floats

    k_zero<<<(MAPN + 1 + 255) / 256, 256, 0, stream>>>(map, MAPN + 1);
    k_prep<<<(B_ * T_ + 255) / 256, 256, 0, stream>>>(targets, map, nvalid, igbuf, rec);
    k_main<<<dim3(BLKS_X, B_), BLK, 0, stream>>>(input, map, igbuf, rec, partials);
    k_final<<<1, BLK, 0, stream>>>(partials, nvalid, out);
}